// MultilabelCrossEntropyLoss_14001593385217
// MI455X (gfx1250) — compile-verified
//
#include <hip/hip_runtime.h>
#include <stdint.h>

#define B_ROWS   16384
#define NLAB     8192
#define XLAB     50
#define TGT_STR  (XLAB + 1)
#define LOG2E    1.4426950408889634f
#define LN2      0.6931471805599453f
#define NEG_FLT_MAX (-3.402823466e38f)

__device__ __forceinline__ float waveMax32(float v) {
#pragma unroll
  for (int off = 16; off > 0; off >>= 1)
    v = fmaxf(v, __shfl_xor(v, off, 32));
  return v;
}
__device__ __forceinline__ float waveSum32(float v) {
#pragma unroll
  for (int off = 16; off > 0; off >>= 1)
    v += __shfl_xor(v, off, 32);
  return v;
}

// One 256-thread workgroup per row. Row (32 KB) is streamed HBM->LDS with
// gfx1250 async-to-LDS ops, then max / sum-exp / gather all read LDS.
__global__ __launch_bounds__(256) void row_loss_kernel(
    const float* __restrict__ prd, const int* __restrict__ tgt,
    float* __restrict__ per_row, float* __restrict__ keep_flag)
{
  __shared__ float smem[NLAB];       // 32 KB row tile
  __shared__ float red_a[8];
  __shared__ float red_b[8];
  __shared__ float bcast[2];

  const int row  = blockIdx.x;
  const int tid  = threadIdx.x;
  const int lane = tid & 31;
  const int wv   = tid >> 5;

  // ---------------- Phase 1: async copy global -> LDS ----------------
  {
    uint64_t gbase    = (uint64_t)(prd + (size_t)row * NLAB);
    unsigned lds_base = (unsigned)(uintptr_t)(&smem[0]);
#pragma unroll
    for (int i = 0; i < 8; ++i) {
      unsigned byteoff = (unsigned)((i * 256 + tid) * 16); // 16 B per lane
      unsigned ldsaddr = lds_base + byteoff;
      asm volatile("global_load_async_to_lds_b128 %0, %1, %2"
                   :: "v"(ldsaddr), "v"(byteoff), "s"(gbase)
                   : "memory");
    }
    asm volatile("s_wait_asynccnt 0" ::: "memory");
  }
  __syncthreads();

  // ---------------- Phase 2: row max (from LDS) ----------------
  const float4* sm4 = (const float4*)smem;
  float m = NEG_FLT_MAX;
#pragma unroll
  for (int i = 0; i < 8; ++i) {
    float4 v = sm4[i * 256 + tid];
    m = fmaxf(m, fmaxf(fmaxf(v.x, v.y), fmaxf(v.z, v.w)));
  }
  m = waveMax32(m);
  if (lane == 0) red_a[wv] = m;
  __syncthreads();
  if (tid < 32) {
    float t = (lane < 8) ? red_a[lane] : NEG_FLT_MAX;
    t = waveMax32(t);
    if (lane == 0) bcast[0] = t;
  }
  __syncthreads();
  m = bcast[0];
  __syncthreads();   // red_a reuse safety

  // ---------------- Phase 3: Z = sum(exp(x - m)) ----------------
  float s = 0.f;
#pragma unroll
  for (int i = 0; i < 8; ++i) {
    float4 v = sm4[i * 256 + tid];
    s += __builtin_amdgcn_exp2f((v.x - m) * LOG2E);
    s += __builtin_amdgcn_exp2f((v.y - m) * LOG2E);
    s += __builtin_amdgcn_exp2f((v.z - m) * LOG2E);
    s += __builtin_amdgcn_exp2f((v.w - m) * LOG2E);
  }
  s = waveSum32(s);
  if (lane == 0) red_a[wv] = s;
  __syncthreads();
  if (tid < 32) {
    float t = (lane < 8) ? red_a[lane] : 0.f;
    t = waveSum32(t);
    if (lane == 0) bcast[1] = t;
  }
  __syncthreads();
  const float Z = bcast[1];
  __syncthreads();   // red_a reuse safety

  // ---------------- Phase 4: gather labels ----------------
  float g = 0.f, ign = 0.f;
  if (tid < XLAB) {
    int lab = tgt[(size_t)row * TGT_STR + tid];
    if (lab == 0) ign = 1.f;                       // IGNORE_IDX == 0
    else          g   = __builtin_amdgcn_exp2f((smem[lab] - m) * LOG2E);
  }
  g   = waveSum32(g);
  ign = waveSum32(ign);
  if (lane == 0) { red_a[wv] = g; red_b[wv] = ign; }
  __syncthreads();

  if (tid == 0) {
    float G = 0.f, I = 0.f;
#pragma unroll
    for (int i = 0; i < 8; ++i) { G += red_a[i]; I += red_b[i]; }
    int   num_listed  = tgt[(size_t)row * TGT_STR + XLAB];
    float num_correct = (float)num_listed - I;
    bool  keep        = num_correct > 0.f;
    float mass        = G / Z;
    // -log(mass + 1e-5)  via v_log_f32 (log2) * ln2
    float per = keep ? -(__builtin_amdgcn_logf(mass + 1e-5f) * LN2) : 0.f;
    per_row[row]   = per;
    keep_flag[row] = keep ? 1.f : 0.f;
  }
}

// Deterministic single-workgroup reduction of 16384 per-row values.
__global__ __launch_bounds__(1024) void finalize_kernel(
    const float* __restrict__ per_row, const float* __restrict__ keep_flag,
    float* __restrict__ out, int n)
{
  __shared__ float ra[32];
  __shared__ float rb[32];
  const int tid = threadIdx.x, lane = tid & 31, wv = tid >> 5;
  float ls = 0.f, ks = 0.f;
  for (int i = tid; i < n; i += 1024) {
    ls += per_row[i];
    ks += keep_flag[i];
  }
  ls = waveSum32(ls);
  ks = waveSum32(ks);
  if (lane == 0) { ra[wv] = ls; rb[wv] = ks; }
  __syncthreads();
  if (tid < 32) {
    float a = ra[lane];
    float b = rb[lane];
    a = waveSum32(a);
    b = waveSum32(b);
    if (tid == 0) out[0] = a / fmaxf(b, 1.0f);
  }
}

extern "C" void kernel_launch(void* const* d_in, const int* in_sizes, int n_in,
                              void* d_out, int out_size, void* d_ws, size_t ws_size,
                              hipStream_t stream) {
  (void)in_sizes; (void)n_in; (void)out_size; (void)ws_size;
  const float* prd = (const float*)d_in[0];
  const int*   tgt = (const int*)d_in[1];
  float* ws   = (float*)d_ws;
  float* per  = ws;            // B_ROWS floats
  float* keep = ws + B_ROWS;   // B_ROWS floats

  row_loss_kernel<<<B_ROWS, 256, 0, stream>>>(prd, tgt, per, keep);
  finalize_kernel<<<1, 1024, 0, stream>>>(per, keep, (float*)d_out, B_ROWS);
}